// BH4_59940563583083
// MI455X (gfx1250) — compile-verified
//
#include <hip/hip_runtime.h>

// BH4 butterfly block-diagonal transform for MI455X (gfx1250, wave32, WMMA).
//
// Roofline: 17.2 GFLOP fp32, ~172 MB HBM (x 33.5 + w 4 + out 134 MB).
// Matmuls run on V_WMMA_F32_16X16X4_F32 (exact fp32 path); the 134 MB output
// stream — the only one near the 23.3 TB/s roofline — is written as coalesced
// non-temporal b128 stores via an LDS-staged epilogue.
//
// vs round 2:
//  * stage 3 now also stores (permuted) to LDS; a coalesced epilogue pass does
//    the 0.7/0.3 blend + bias and b128 global stores (round 2 scattered 4B
//    stores across 64B-strided lanes -> 16 dword stores per output line).
//  * skewed LDS column mapping f -> f + 4*(f>>4): permuted stores go from
//    8-way to 2-way bank conflicts while keeping v2f/v4f alignment.
//  * two independent WMMA accumulators halve the RAW-on-C chain depth.

typedef __attribute__((ext_vector_type(2))) float v2f;
typedef __attribute__((ext_vector_type(4))) float v4f;
typedef __attribute__((ext_vector_type(8))) float v8f;

#define D_IN   1024        // input feature dim
#define NB     16          // blocks per stage
#define BS     64          // block size
#define NREP   4           // repeats
#define TR     16          // rows per workgroup tile
#define SW     1276        // skewed LDS row stride (floats); max col 1275
#define OUTD   4096
#define DECAY  0.7f

#define STAGE_FLOATS (NB * BS * BS)                             // 65536 per (r,s)
#define WT_BYTES     (NREP * 4 * STAGE_FLOATS * sizeof(float))  // 4 MB packed

// skewed physical column for logical feature f: f + 4*(f>>4)
__device__ __forceinline__ int skewcol(int f) { return f + ((f >> 4) << 2); }

// ---------------------------------------------------------------------------
// One-shot weight pack: w[r][s][nb][i][o] -> wt[tile][lane][kb][c]
// tile = ((r*4+s)*16+nb)*4 + j;  B element (k = kb*4 + (lane>>4)*2 + c,
// n = j*16 + (lane&15)).  Per lane 128 B contiguous -> compute kernel fetches
// a full B tile as 8 coalesced global_load_b128 (weights stay L2-resident).
// ---------------------------------------------------------------------------
__global__ __launch_bounds__(256) void bh4_wt_pack(const float* __restrict__ w,
                                                   float* __restrict__ wt) {
    const int tile   = blockIdx.x;       // 0..1023
    const int tid    = threadIdx.x;      // 0..255
    const int j      = tile & 3;
    const int rsnb   = tile >> 2;        // (r*4+s)*16 + nb
    const int lane32 = tid >> 3;         // emulated wmma lane 0..31
    const int kb0    = (tid & 7) * 2;    // kb pair handled by this thread
    const int hs     = lane32 >> 4;
    const int lh     = lane32 & 15;
    const int o      = j * 16 + lh;

    const float* src = w + (size_t)rsnb * (BS * BS) + o;
    v4f q;
    q.x = src[(kb0 * 4 + hs * 2 + 0) * BS];
    q.y = src[(kb0 * 4 + hs * 2 + 1) * BS];
    q.z = src[((kb0 + 1) * 4 + hs * 2 + 0) * BS];
    q.w = src[((kb0 + 1) * 4 + hs * 2 + 1) * BS];
    *(v4f*)&wt[(size_t)tile * 1024 + lane32 * 32 + (tid & 7) * 4] = q;
}

// ---------------------------------------------------------------------------
// One 16x16 output tile of one block-diagonal matmul (K = 64 = 16 wmma steps).
// Operands preloaded to VGPRs; two accumulators halve the RAW chain depth.
// ---------------------------------------------------------------------------
template <bool USE_WT>
__device__ __forceinline__ v8f bh4_tile(const float* __restrict__ in,
                                        const float* __restrict__ wstage,
                                        int nb, int j, int lane, int lh, int hs) {
    // A fragment (m=lh, k = kb*4 + hs*2 + c) at skewed column:
    // col = nb*64 + kb*4 + hs*2 + 4*(nb*4 + (kb>>2)) = nb*80 + kb*4 + (kb>>2)*4 + hs*2
    const float* arow = in + lh * SW + nb * 80 + hs * 2;
    v2f af[16];
#pragma unroll
    for (int kb = 0; kb < 16; ++kb)
        af[kb] = *(const v2f*)(arow + kb * 4 + ((kb >> 2) << 2));  // ds_load_b64 x16

    v8f acc0 = {}, acc1 = {};
    if (USE_WT) {
        const float* bt = wstage + ((size_t)nb * 4 + j) * 1024 + lane * 32;
        v4f bq[8];
#pragma unroll
        for (int i = 0; i < 8; ++i)
            bq[i] = *(const v4f*)(bt + i * 4);                     // global_load_b128 x8
#pragma unroll
        for (int i = 0; i < 8; ++i) {
            v2f blo = __builtin_shufflevector(bq[i], bq[i], 0, 1);
            v2f bhi = __builtin_shufflevector(bq[i], bq[i], 2, 3);
            acc0 = __builtin_amdgcn_wmma_f32_16x16x4_f32(
                       false, af[2 * i],     false, blo, (short)0, acc0, false, false);
            acc1 = __builtin_amdgcn_wmma_f32_16x16x4_f32(
                       false, af[2 * i + 1], false, bhi, (short)0, acc1, false, false);
        }
    } else {
        // fallback: fragment gather from the natural [i][o] weight layout
        const float* wblk = wstage + nb * (BS * BS) + (hs * 2) * BS + j * 16 + lh;
        v2f bf[16];
#pragma unroll
        for (int kb = 0; kb < 16; ++kb) {
            bf[kb].x = wblk[kb * 4 * BS];
            bf[kb].y = wblk[kb * 4 * BS + BS];
        }
#pragma unroll
        for (int kb = 0; kb < 16; ++kb) {
            v8f& a = (kb & 1) ? acc1 : acc0;
            a = __builtin_amdgcn_wmma_f32_16x16x4_f32(
                    false, af[kb], false, bf[kb], (short)0, a, false, false);
        }
    }
    return acc0 + acc1;
}

// ---------------------------------------------------------------------------
// Main kernel: 16-row tile x one repeat per workgroup, 8 wave32 waves.
// 4 stages of (block-diag matmul -> butterfly permute) all LDS ping-pong,
// then a coalesced epilogue: blend + bias + non-temporal b128 global stores.
// ---------------------------------------------------------------------------
template <bool USE_WT>
__global__ __launch_bounds__(256) void bh4_wmma_kernel(
    const float* __restrict__ x,      // [N, 1024]
    const float* __restrict__ wsrc,   // packed wt or raw w (same stage stride)
    const float* __restrict__ bias,   // [4096]
    float* __restrict__ out)          // [N, 4096]
{
    __shared__ float lds[2][TR * SW];   // 2 x 16 x 1276 x 4B = 163328 B (2 WG/WGP)

    const int tid  = threadIdx.x;
    const int lane = tid & 31;
    const int wv   = tid >> 5;
    const int lh   = lane & 15;
    const int hs   = lane >> 4;

    const int rowBase = blockIdx.x * TR;
    const int r       = blockIdx.y;

    // cooperative 16x1024 x-tile load into lds[0] (skewed columns)
    {
        const float* xt = x + (size_t)rowBase * D_IN;
#pragma unroll
        for (int it = 0; it < 16; ++it) {
            const int row = it;              // 256 threads cover one 1024-f row
            const int c4  = tid;             // float4 chunk within row
            *(v4f*)&lds[0][row * SW + skewcol(c4 * 4)] =
                *(const v4f*)&xt[(size_t)row * D_IN + c4 * 4];
        }
    }
    __syncthreads();

    // 4 stages: block-diag matmul, butterfly permute folded into skewed store
    for (int s = 0; s < 4; ++s) {
        const float* __restrict__ in     = lds[s & 1];
        float* __restrict__       outb   = lds[(s + 1) & 1];
        const float* __restrict__ wstage = wsrc + (size_t)(r * 4 + s) * STAGE_FLOATS;

        if (s < 3)  // warm next stage's weights (global_prefetch_b8)
            __builtin_prefetch(wsrc + (size_t)(r * 4 + s + 1) * STAGE_FLOATS + wv * 8192, 0, 1);

#pragma unroll
        for (int t = 0; t < 8; ++t) {
            const int p  = wv * 8 + t;       // (block, ntile) pair 0..63
            const int nb = p >> 2;
            const int j  = p & 3;
            v8f acc = bh4_tile<USE_WT>(in, wstage, nb, j, lane, lh, hs);

            // butterfly: (nb, o) -> feature o*16+nb; skewed col = o*20 + nb
            const int colp = (j * 16 + lh) * 20 + nb;
#pragma unroll
            for (int v = 0; v < 8; ++v)
                outb[(v + hs * 8) * SW + colp] = acc[v];
        }
        __syncthreads();
    }

    // epilogue: y in lds[0]; fused 0.7*y + 0.3*x + bias, coalesced b128 stores
    {
        const float* __restrict__ xt = x + (size_t)rowBase * D_IN;
#pragma unroll
        for (int it = 0; it < 16; ++it) {
            const int row = it;
            const int c4  = tid;
            const int f   = c4 * 4;
            v4f y  = *(const v4f*)&lds[0][row * SW + skewcol(f)];
            v4f xv = *(const v4f*)&xt[(size_t)row * D_IN + f];
            v4f bv = *(const v4f*)&bias[r * D_IN + f];
            v4f o  = DECAY * y + (1.0f - DECAY) * xv + bv;
            // output never re-read -> non-temporal, keep L2 for x/weights
            __builtin_nontemporal_store(
                o, (v4f*)&out[(size_t)(rowBase + row) * OUTD + r * D_IN + f]);
        }
    }
}

extern "C" void kernel_launch(void* const* d_in, const int* in_sizes, int n_in,
                              void* d_out, int out_size, void* d_ws, size_t ws_size,
                              hipStream_t stream) {
    const float* x    = (const float*)d_in[0];   // [B,S,1024] fp32
    const float* w    = (const float*)d_in[1];   // [4,4,16,64,64] fp32
    const float* bias = (const float*)d_in[2];   // [4096] fp32
    float*       out  = (float*)d_out;           // [B,S,4096] fp32

    const int nrows = in_sizes[0] / D_IN;        // 8192
    dim3 grid(nrows / TR, NREP);                 // (512, 4)

    if (ws_size >= WT_BYTES) {
        float* wt = (float*)d_ws;
        bh4_wt_pack<<<NREP * 4 * NB * 4, 256, 0, stream>>>(w, wt);
        bh4_wmma_kernel<true><<<grid, 256, 0, stream>>>(x, wt, bias, out);
    } else {
        bh4_wmma_kernel<false><<<grid, 256, 0, stream>>>(x, w, bias, out);
    }
}